// ProjectionAttentionKernel_44753559225153
// MI455X (gfx1250) — compile-verified
//
#include <hip/hip_runtime.h>

typedef __attribute__((ext_vector_type(2))) float v2f;
typedef __attribute__((ext_vector_type(8))) float v8f;

#define DIM_C   768
#define HEADS_C 12
#define HD_C    64
#define B_C     8
#define N_C     512
#define BN_C    (B_C * N_C)             // 4096
#define QKVD_C  (3 * DIM_C)             // 2304
#define BH_C    (B_C * HEADS_C)         // 96
#define HNN_C   (HEADS_C * N_C * N_C)   // 3145728
#define LDA     68                      // padded LDS row stride (floats): conflict-free, 16B aligned

// workspace float offsets
#define WS_QKV     0
#define WS_NQ      (BN_C * QKVD_C)                   // 9437184
#define WS_NK      (WS_NQ + BH_C * N_C)              // +49152
#define WS_OUTPRE  (WS_NK + BH_C * N_C)              // +49152
#define WS_ACCUM_F (WS_OUTPRE + BH_C * N_C * HD_C)   // 12681216 (byte offset 8-aligned)

__device__ __forceinline__ v8f wmma4(v2f a, v2f b, v8f c) {
    // V_WMMA_F32_16X16X4_F32: D = A(16x4) * B(4x16) + C(16x16), all fp32
    return __builtin_amdgcn_wmma_f32_16x16x4_f32(
        false, a, false, b, (short)0, c, false, false);
}

// Coalesced async copy of one 16B chunk: global -> LDS (ASYNCcnt-tracked)
__device__ __forceinline__ void async_b128(float* lds_dst, const float* gsrc) {
    const unsigned laddr = (unsigned)(size_t)lds_dst;   // low 32 bits of generic ptr = LDS offset
    asm volatile("global_load_async_to_lds_b128 %0, %1, off"
                 :: "v"(laddr), "v"(gsrc) : "memory");
}
__device__ __forceinline__ void async_wait0() {
    asm volatile("s_wait_asynccnt 0x0" ::: "memory");
}

// ---------------- LDS-staged 128x64 GEMM core ----------------
// Block = 8 waves; wave wv owns rows [wv*16, wv*16+16) of the 128-row panel.
// BT=false: fragment B[k][n] read as sB[col][k]  (sB rows = output columns)
// BT=true : fragment B[k][n] read as sB[k][col]  (sB rows = K values)
template <bool BT, typename GA, typename GB>
__device__ __forceinline__ void staged_gemm_core(int K, GA ga, GB gb,
                                                 float* sA, float* sB, v8f (&acc)[4]) {
    const int t = threadIdx.x;
    const int lane = t & 31, half = lane >> 4, l16 = lane & 15, wv = t >> 5;
    const int lrow = wv * 16 + l16;
    for (int kc = 0; kc < K; kc += 64) {
        // stage A panel: 128 rows x 64 floats (2048 x b128 chunks, 8 per thread)
#pragma unroll
        for (int i = 0; i < 8; ++i) {
            const int c = i * 256 + t, rr = c >> 4, o4 = (c & 15) * 4;
            async_b128(sA + rr * LDA + o4, ga(rr, kc, o4));
        }
        // stage B panel: 64 rows x 64 floats (1024 x b128 chunks, 4 per thread)
#pragma unroll
        for (int i = 0; i < 4; ++i) {
            const int c = i * 256 + t, rr = c >> 4, o4 = (c & 15) * 4;
            async_b128(sB + rr * LDA + o4, gb(rr, kc, o4));
        }
        async_wait0();
        __syncthreads();
#pragma unroll
        for (int k0 = 0; k0 < 64; k0 += 4) {
            const int kk = k0 + 2 * half;
            v2f a; a.x = sA[lrow * LDA + kk]; a.y = sA[lrow * LDA + kk + 1];
#pragma unroll
            for (int j = 0; j < 4; ++j) {
                const int col = j * 16 + l16;
                v2f b;
                if (BT) { b.x = sB[kk * LDA + col];  b.y = sB[(kk + 1) * LDA + col]; }
                else    { b.x = sB[col * LDA + kk];  b.y = sB[col * LDA + kk + 1]; }
                acc[j] = wmma4(a, b, acc[j]);
            }
        }
        __syncthreads();   // protect LDS before next chunk's async writes
    }
}

// ---------------- 0: zero LN accumulators ----------------
__global__ void k_zero_accum(double* __restrict__ accum) {
    if (threadIdx.x < 16) accum[threadIdx.x] = 0.0;
}

// ---------------- 1: qkv = x @ qkv_w^T  (4096x768 * 768->2304) ----------------
__global__ __launch_bounds__(256) void k_qkv_gemm(const float* __restrict__ x,
                                                  const float* __restrict__ w,
                                                  float* __restrict__ qkv) {
    __shared__ __attribute__((aligned(16))) float sA[128 * LDA];
    __shared__ __attribute__((aligned(16))) float sB[64 * LDA];
    const int blk = blockIdx.x;
    const int mg = blk / 36, nt = blk % 36;       // 32 row-groups x 36 col-tiles
    const int row0 = mg * 128, col0 = nt * 64;
    v8f acc[4] = {};
    staged_gemm_core<false>(DIM_C,
        [&](int rr, int kc, int o4) { return x + (size_t)(row0 + rr) * DIM_C + kc + o4; },
        [&](int rr, int kc, int o4) { return w + (size_t)(col0 + rr) * DIM_C + kc + o4; },
        sA, sB, acc);
    const int lane = threadIdx.x & 31, half = lane >> 4, l16 = lane & 15, wv = threadIdx.x >> 5;
#pragma unroll
    for (int j = 0; j < 4; ++j) {
        const int col = col0 + j * 16 + l16;
#pragma unroll
        for (int i = 0; i < 8; ++i) {
            const int row = row0 + wv * 16 + i + 8 * half;
            qkv[(size_t)row * QKVD_C + col] = acc[j][i];
        }
    }
}

// ---------------- 2: inv squared norms of q and k rows ----------------
__global__ void k_norms(const float* __restrict__ qkv, float* __restrict__ invq,
                        float* __restrict__ invk) {
    const int lane = threadIdx.x & 31;
    int wave = (int)((blockIdx.x * blockDim.x + threadIdx.x) >> 5);
    const int total = BH_C * N_C;                 // 49152 rows each for q, k
    int sel = 0;
    if (wave >= total) { sel = 1; wave -= total; }
    const int bh = wave / N_C, n = wave % N_C;
    const int b = bh / HEADS_C, h = bh % HEADS_C;
    const float* src = qkv + (size_t)(b * N_C + n) * QKVD_C + sel * DIM_C + h * HD_C;
    float v0 = src[lane], v1 = src[lane + 32];
    float s = v0 * v0 + v1 * v1;
#pragma unroll
    for (int m = 16; m >= 1; m >>= 1) s += __shfl_xor(s, m, 32);
    if (lane == 0) (sel ? invk : invq)[wave] = 1.0f / s;
}

// ---------------- 3: raw attn = (q^2/|q|^2) @ (k^2/|k|^2)^T + LN stats ----------------
__global__ __launch_bounds__(256) void k_attn(const float* __restrict__ qkv,
                                              const float* __restrict__ invq,
                                              const float* __restrict__ invk,
                                              float* __restrict__ attn,
                                              double* __restrict__ accum) {
    __shared__ __attribute__((aligned(16))) float sA[128 * LDA];   // raw q rows
    __shared__ __attribute__((aligned(16))) float sB[64 * LDA];    // raw k rows
    const int t = threadIdx.x, lane = t & 31, half = lane >> 4, l16 = lane & 15, wv = t >> 5;
    const int blk = blockIdx.x;
    const int bh = blk >> 5;                      // 32 blocks per (b,h)
    const int r = blk & 31, nt = r >> 2, mg = r & 3;
    const int b = bh / HEADS_C, h = bh % HEADS_C;
    const int row0 = mg * 128, col0 = nt * 64;
    const float* qbase = qkv + (size_t)(b * N_C) * QKVD_C + h * HD_C;   // q[n][d] @ qbase[n*2304+d]
    const float* kbase = qbase + DIM_C;
    // async stage: 128 q rows + 64 k rows (each row 256B contiguous)
#pragma unroll
    for (int i = 0; i < 8; ++i) {
        const int c = i * 256 + t, rr = c >> 4, o4 = (c & 15) * 4;
        async_b128(sA + rr * LDA + o4, qbase + (size_t)(row0 + rr) * QKVD_C + o4);
    }
#pragma unroll
    for (int i = 0; i < 4; ++i) {
        const int c = i * 256 + t, rr = c >> 4, o4 = (c & 15) * 4;
        async_b128(sB + rr * LDA + o4, kbase + (size_t)(col0 + rr) * QKVD_C + o4);
    }
    async_wait0();
    __syncthreads();

    const int lrow = wv * 16 + l16;
    const float sa = invq[bh * N_C + row0 + lrow];
    float sb[4];
#pragma unroll
    for (int j = 0; j < 4; ++j) sb[j] = invk[bh * N_C + col0 + j * 16 + l16];
    v8f acc[4] = {};
#pragma unroll
    for (int k0 = 0; k0 < HD_C; k0 += 4) {
        const int kk = k0 + 2 * half;
        const float q0 = sA[lrow * LDA + kk], q1 = sA[lrow * LDA + kk + 1];
        v2f a; a.x = q0 * q0 * sa; a.y = q1 * q1 * sa;
#pragma unroll
        for (int j = 0; j < 4; ++j) {
            const int lc = j * 16 + l16;
            const float c0 = sB[lc * LDA + kk], c1 = sB[lc * LDA + kk + 1];
            v2f bb; bb.x = c0 * c0 * sb[j]; bb.y = c1 * c1 * sb[j];
            acc[j] = wmma4(a, bb, acc[j]);
        }
    }
    // write raw attn (into d_out attn region) + accumulate LN stats
    float* abase = attn + (size_t)bh * N_C * N_C;
    double s = 0.0, s2 = 0.0;
#pragma unroll
    for (int j = 0; j < 4; ++j) {
        const int col = col0 + j * 16 + l16;
#pragma unroll
        for (int i = 0; i < 8; ++i) {
            const int row = row0 + wv * 16 + i + 8 * half;
            const float vv = acc[j][i];
            abase[(size_t)row * N_C + col] = vv;
            s += vv; s2 += (double)vv * (double)vv;
        }
    }
#pragma unroll
    for (int m = 16; m >= 1; m >>= 1) { s += __shfl_xor(s, m, 32); s2 += __shfl_xor(s2, m, 32); }
    if (lane == 0) {
        atomicAdd(&accum[2 * b + 0], s);
        atomicAdd(&accum[2 * b + 1], s2);
    }
}

// ---------------- 4: in-place LayerNorm over d_out attn region ----------------
__global__ void k_ln(float* __restrict__ attn, const float* __restrict__ lw,
                     const float* __restrict__ lb, const double* __restrict__ accum) {
    const size_t total = (size_t)B_C * HNN_C;
    const size_t stride = (size_t)gridDim.x * blockDim.x;
    for (size_t idx = (size_t)blockIdx.x * blockDim.x + threadIdx.x; idx < total; idx += stride) {
        const int b = (int)(idx / HNN_C);
        const int r = (int)(idx - (size_t)b * HNN_C);
        const double cnt = (double)HNN_C;
        const double mean = accum[2 * b] / cnt;
        const double var = accum[2 * b + 1] / cnt - mean * mean;
        const float rstd = (float)(1.0 / sqrt(var + 1e-5));
        const float a = attn[idx];
        attn[idx] = (a - (float)mean) * rstd * lw[r] + lb[r];
    }
}

// ---------------- 5: out_pre = attn @ v  (96 x [512x512 * 512x64]) ----------------
__global__ __launch_bounds__(256) void k_av(const float* __restrict__ attn,
                                            const float* __restrict__ qkv,
                                            float* __restrict__ outpre) {
    __shared__ __attribute__((aligned(16))) float sA[128 * LDA];
    __shared__ __attribute__((aligned(16))) float sB[64 * LDA];
    const int blk = blockIdx.x;
    const int bh = blk >> 2, mg = blk & 3;        // 4 row-groups per (b,h)
    const int b = bh / HEADS_C, h = bh % HEADS_C;
    const int row0 = mg * 128;
    const float* abase = attn + (size_t)bh * N_C * N_C;
    const float* vbase = qkv + (size_t)(b * N_C) * QKVD_C + 2 * DIM_C + h * HD_C; // v[k][d]
    v8f acc[4] = {};
    staged_gemm_core<true>(N_C,
        [&](int rr, int kc, int o4) { return abase + (size_t)(row0 + rr) * N_C + kc + o4; },
        [&](int rr, int kc, int o4) { return vbase + (size_t)(kc + rr) * QKVD_C + o4; },
        sA, sB, acc);
    const int lane = threadIdx.x & 31, half = lane >> 4, l16 = lane & 15, wv = threadIdx.x >> 5;
#pragma unroll
    for (int j = 0; j < 4; ++j) {
        const int col = j * 16 + l16;
#pragma unroll
        for (int i = 0; i < 8; ++i) {
            const int row = row0 + wv * 16 + i + 8 * half;
            outpre[((size_t)bh * N_C + row) * HD_C + col] = acc[j][i];
        }
    }
}

// ---------------- 6: out = gather(out_pre) @ proj_w^T + proj_b ----------------
__global__ __launch_bounds__(256) void k_proj(const float* __restrict__ outpre,
                                              const float* __restrict__ pw,
                                              const float* __restrict__ pb,
                                              float* __restrict__ out) {
    __shared__ __attribute__((aligned(16))) float sA[128 * LDA];
    __shared__ __attribute__((aligned(16))) float sB[64 * LDA];
    const int blk = blockIdx.x;
    const int mg = blk / 12, nt = blk % 12;       // 32 row-groups x 12 col-tiles
    const int row0 = mg * 128, col0 = nt * 64;
    v8f acc[4] = {};
    staged_gemm_core<false>(DIM_C,
        [&](int rr, int kc, int o4) {
            const int row = row0 + rr, bb = row >> 9, nn = row & 511;
            const int kk = kc + o4, h = kk >> 6, d = kk & 63;  // chunk stays inside one head
            return outpre + (((size_t)(bb * HEADS_C + h) * N_C + nn) * HD_C + d);
        },
        [&](int rr, int kc, int o4) { return pw + (size_t)(col0 + rr) * DIM_C + kc + o4; },
        sA, sB, acc);
    const int lane = threadIdx.x & 31, half = lane >> 4, l16 = lane & 15, wv = threadIdx.x >> 5;
#pragma unroll
    for (int j = 0; j < 4; ++j) {
        const int col = col0 + j * 16 + l16;
        const float bias = pb[col];
#pragma unroll
        for (int i = 0; i < 8; ++i) {
            const int row = row0 + wv * 16 + i + 8 * half;
            out[(size_t)row * DIM_C + col] = acc[j][i] + bias;
        }
    }
}

extern "C" void kernel_launch(void* const* d_in, const int* in_sizes, int n_in,
                              void* d_out, int out_size, void* d_ws, size_t ws_size,
                              hipStream_t stream) {
    (void)in_sizes; (void)n_in; (void)out_size; (void)ws_size;
    const float* x  = (const float*)d_in[0];
    const float* qw = (const float*)d_in[1];
    const float* lw = (const float*)d_in[2];
    const float* lb = (const float*)d_in[3];
    const float* pw = (const float*)d_in[4];
    const float* pb = (const float*)d_in[5];

    float* out  = (float*)d_out;                       // (B,N,DIM) = 3,145,728 floats
    float* attn = out + (size_t)BN_C * DIM_C;          // (B,H,N,N) = 25,165,824 floats

    float*  ws     = (float*)d_ws;
    float*  qkv    = ws + WS_QKV;
    float*  invq   = ws + WS_NQ;
    float*  invk   = ws + WS_NK;
    float*  outpre = ws + WS_OUTPRE;
    double* accum  = (double*)(ws + WS_ACCUM_F);

    k_zero_accum<<<1, 32, 0, stream>>>(accum);
    k_qkv_gemm<<<32 * 36, 256, 0, stream>>>(x, qw, qkv);       // 1152 blocks
    k_norms<<<12288, 256, 0, stream>>>(qkv, invq, invk);       // 98304 wave rows
    k_attn<<<96 * 32, 256, 0, stream>>>(qkv, invq, invk, attn, accum); // 3072 blocks
    k_ln<<<4096, 256, 0, stream>>>(attn, lw, lb, accum);
    k_av<<<96 * 4, 256, 0, stream>>>(attn, qkv, outpre);       // 384 blocks
    k_proj<<<32 * 12, 256, 0, stream>>>(outpre, pw, pb, out);  // 384 blocks
}